// BlockWiseEmbedding_31731218383117
// MI455X (gfx1250) — compile-verified
//
#include <hip/hip_runtime.h>

// 16-byte vector type -> global/ds b128 ops
typedef __attribute__((ext_vector_type(4))) float v4f;

#define WAVES_PER_BLOCK 8
#define TOK_PER_WAVE    4
#define THREADS         (WAVES_PER_BLOCK * 32)
#define TOK_PER_BLOCK   (WAVES_PER_BLOCK * TOK_PER_WAVE)
#define F4_PER_ROW      128   // DIM=512 -> 128 float4 per embedding row

// ---- CDNA5 async data-mover: HBM -> LDS, tracked by ASYNCcnt (ISA 15.18.3 op 98)
__device__ __forceinline__ void async_gather_b128(unsigned lds_addr, const void* gaddr) {
    asm volatile("global_load_async_to_lds_b128 %0, %1, off"
                 :: "v"(lds_addr), "v"(gaddr)
                 : "memory");
}
__device__ __forceinline__ void wait_async_0() {
    asm volatile("s_wait_asynccnt 0x0" ::: "memory");
}

// Branchless table-base select: integer ?: lowers to v_cndmask (no EXEC branches).
__device__ __forceinline__ const v4f* pick_table(int blk, const float* t0, const float* t1,
                                                 const float* t2, const float* t3) {
    unsigned long long p = (unsigned long long)t0;
    p = (blk == 1) ? (unsigned long long)t1 : p;
    p = (blk == 2) ? (unsigned long long)t2 : p;
    p = (blk == 3) ? (unsigned long long)t3 : p;
    return (const v4f*)p;
}

__global__ __launch_bounds__(THREADS) void emb_gather_async(
    const int* __restrict__ src,
    const int* __restrict__ blockAssign,
    const int* __restrict__ localAssign,
    const float* __restrict__ t0, const float* __restrict__ t1,
    const float* __restrict__ t2, const float* __restrict__ t3,
    v4f* __restrict__ out,
    int nTokens)
{
    __shared__ v4f smem[WAVES_PER_BLOCK * TOK_PER_WAVE * F4_PER_ROW]; // 64 KiB/block

    const int wave    = threadIdx.x >> 5;
    const int lane    = threadIdx.x & 31;
    const int tokBase = (blockIdx.x * WAVES_PER_BLOCK + wave) * TOK_PER_WAVE;

    v4f* mybuf = &smem[wave * TOK_PER_WAVE * F4_PER_ROW];       // 8 KiB per wave
    const unsigned ldsBase = (unsigned)(unsigned long long)(void*)mybuf;

    if (tokBase + TOK_PER_WAVE <= nTokens) {   // wave-uniform: EXEC all-ones inside
        // Resolve the 4 row pointers (token -> block/local -> row base).
        const v4f* rows[TOK_PER_WAVE];
#pragma unroll
        for (int t = 0; t < TOK_PER_WAVE; ++t) {
            const int idx = src[tokBase + t];
            const int blk = blockAssign[idx];
            const long loc = localAssign[idx];
            rows[t] = pick_table(blk, t0, t1, t2, t3) + loc * F4_PER_ROW;
        }
        // Phase A: 16 async b128 gathers (4 per 2KiB row) -> 8 KiB in flight/wave.
#pragma unroll
        for (int t = 0; t < TOK_PER_WAVE; ++t) {
#pragma unroll
            for (int k = 0; k < 4; ++k) {
                const int e = k * 32 + lane;                    // float4 column 0..127
                async_gather_b128(ldsBase + (unsigned)((t * F4_PER_ROW + e) * 16),
                                  (const void*)(rows[t] + e));
            }
        }
        // Single drain of the async engine for the whole 8 KiB batch.
        wait_async_0();
        // Phase B: LDS -> VGPR (ds_load_b128) -> NT global_store_b128.
        // NT keeps the 128MB write-once output out of L2 so the ~200MB of
        // tables stay resident in the 192MB L2 across graph replays.
#pragma unroll
        for (int t = 0; t < TOK_PER_WAVE; ++t) {
            v4f* dst = out + (long)(tokBase + t) * F4_PER_ROW;
#pragma unroll
            for (int k = 0; k < 4; ++k) {
                const int e = k * 32 + lane;
                const v4f v = mybuf[t * F4_PER_ROW + e];
                __builtin_nontemporal_store(v, dst + e);
            }
        }
    } else {
        // Tail (not hit for 65536 tokens): plain guarded per-token path.
        for (int t = 0; t < TOK_PER_WAVE; ++t) {
            const int tok = tokBase + t;
            if (tok < nTokens) {
                const int idx = src[tok];
                const int blk = blockAssign[idx];
                const long loc = localAssign[idx];
                const v4f* row = pick_table(blk, t0, t1, t2, t3) + loc * F4_PER_ROW;
                v4f* dst = out + (long)tok * F4_PER_ROW;
                for (int k = 0; k < 4; ++k) {
                    const int e = k * 32 + lane;
                    __builtin_nontemporal_store(row[e], dst + e);
                }
            }
        }
    }
}

// Generic fallback for any dim != 512 (not expected with this harness).
__global__ __launch_bounds__(256) void emb_gather_generic(
    const int* __restrict__ src,
    const int* __restrict__ blockAssign,
    const int* __restrict__ localAssign,
    const float* __restrict__ t0, const float* __restrict__ t1,
    const float* __restrict__ t2, const float* __restrict__ t3,
    float* __restrict__ out,
    int dim, long total)
{
    const long stride = (long)gridDim.x * blockDim.x;
    for (long i = (long)blockIdx.x * blockDim.x + threadIdx.x; i < total; i += stride) {
        const int tok = (int)(i / dim);
        const int col = (int)(i % dim);
        const int idx = src[tok];
        const int blk = blockAssign[idx];
        const long loc = localAssign[idx];
        const float* row = (const float*)pick_table(blk, t0, t1, t2, t3) + loc * dim;
        __builtin_nontemporal_store(row[col], &out[i]);
    }
}

extern "C" void kernel_launch(void* const* d_in, const int* in_sizes, int n_in,
                              void* d_out, int out_size, void* d_ws, size_t ws_size,
                              hipStream_t stream) {
    const int*   src = (const int*)d_in[0];
    const int*   ba  = (const int*)d_in[1];
    const int*   la  = (const int*)d_in[2];
    const float* t0  = (const float*)d_in[3];
    const float* t1  = (const float*)d_in[4];
    const float* t2  = (const float*)d_in[5];
    const float* t3  = (const float*)d_in[6];

    const int nTokens = in_sizes[0];          // 65536
    const int dim     = out_size / nTokens;   // 512

    if (dim == 4 * F4_PER_ROW) {
        const int blocks = (nTokens + TOK_PER_BLOCK - 1) / TOK_PER_BLOCK; // 2048
        emb_gather_async<<<blocks, THREADS, 0, stream>>>(
            src, ba, la, t0, t1, t2, t3, (v4f*)d_out, nTokens);
    } else {
        const long total = (long)nTokens * dim;
        const int blocks = (int)((total / 4 + 255) / 256);
        emb_gather_generic<<<blocks, 256, 0, stream>>>(
            src, ba, la, t0, t1, t2, t3, (float*)d_out, dim, total);
    }
}